// QModel_65481071409187
// MI455X (gfx1250) — compile-verified
//
#include <hip/hip_runtime.h>
#include <math.h>

#define NW   18
#define DIM  (1 << NW)          // 262144 amplitudes
#define GROUPS 6                // 6 groups of 3 wires per layer
#define COLS (1 << 15)          // outer indices per group
#define TILES (COLS / 16)       // 2048 wmma tiles per group
#define TILES_PER_WAVE 4        // A-matrix reuse factor

typedef float v2f __attribute__((ext_vector_type(2)));
typedef float v8f __attribute__((ext_vector_type(8)));

struct c32 { float x, y; };
__device__ __forceinline__ c32 cmul(c32 a, c32 b) {
    return { a.x * b.x - a.y * b.y, a.x * b.y + a.y * b.x };
}
__device__ __forceinline__ c32 cadd(c32 a, c32 b) { return { a.x + b.x, a.y + b.y }; }

// ---------------------------------------------------------------------------
// Init: |0...0> state, zero the expval accumulators.
// ---------------------------------------------------------------------------
__global__ void qk_init(float* __restrict__ re, float* __restrict__ im,
                        float* __restrict__ expvals) {
    unsigned i = blockIdx.x * blockDim.x + threadIdx.x;
    re[i] = (i == 0) ? 1.0f : 0.0f;
    im[i] = 0.0f;
    if (i < NW) expvals[i] = 0.0f;
}

// ---------------------------------------------------------------------------
// Build fused gates: for each (layer d, group g) compute the 8x8 complex
// G = U_{3g} (x) U_{3g+1} (x) U_{3g+2}, with U_w = Rz*Ry*Rx, and store the
// 16x16 real matrix [[Gr,-Gi],[Gi,Gr]] row-major. Grid = 12 blocks, 256 thr.
// ---------------------------------------------------------------------------
__global__ void qk_build_gates(const float* __restrict__ params,
                               float* __restrict__ gates) {
    int dg = blockIdx.x;            // 0..11  (= d*6 + g)
    int d  = dg / GROUPS;
    int g  = dg % GROUPS;
    int tid = threadIdx.x;          // 0..255 -> (r,c) of 16x16
    int r = tid >> 4, c = tid & 15;

    c32 U[3][2][2];
    for (int q = 0; q < 3; ++q) {
        int w = 3 * g + q;
        const float* pp = params + (d * NW + w) * 3;
        float cx = cosf(0.5f * pp[0]), sx = sinf(0.5f * pp[0]);
        float cy = cosf(0.5f * pp[1]), sy = sinf(0.5f * pp[1]);
        float cz = cosf(0.5f * pp[2]), sz = sinf(0.5f * pp[2]);
        c32 RX[2][2] = { { {cx, 0.f}, {0.f, -sx} }, { {0.f, -sx}, {cx, 0.f} } };
        c32 RY[2][2] = { { {cy, 0.f}, {-sy, 0.f} }, { {sy, 0.f},  {cy, 0.f} } };
        c32 RZ[2][2] = { { {cz, -sz}, {0.f, 0.f} }, { {0.f, 0.f}, {cz, sz} } };
        c32 T[2][2];
        for (int i = 0; i < 2; ++i)
            for (int j = 0; j < 2; ++j)
                T[i][j] = cadd(cmul(RY[i][0], RX[0][j]), cmul(RY[i][1], RX[1][j]));
        for (int i = 0; i < 2; ++i)
            for (int j = 0; j < 2; ++j)
                U[q][i][j] = cadd(cmul(RZ[i][0], T[0][j]), cmul(RZ[i][1], T[1][j]));
    }
    int r8 = r & 7, c8 = c & 7;
    c32 gv = cmul(cmul(U[0][(r8 >> 2) & 1][(c8 >> 2) & 1],
                       U[1][(r8 >> 1) & 1][(c8 >> 1) & 1]),
                  U[2][r8 & 1][c8 & 1]);
    float val = (r < 8) ? ((c < 8) ? gv.x : -gv.y)
                        : ((c < 8) ? gv.y :  gv.x);
    gates[dg * 256 + r * 16 + c] = val;
}

// ---------------------------------------------------------------------------
// Apply one fused 3-wire group gate in-place via V_WMMA_F32_16X16X4_F32.
// Group bits occupy flat-index bits [p+2 : p]  (p = 15 - 3g).
// Each wave loads the 16x16 real gate (A) once and applies it to
// TILES_PER_WAVE consecutive 16-column tiles:
//   D(16x16) = sum_{c4=0..3} A(16x4) * B(4x16)   (16 static v_wmma / wave)
// Grid = 64 blocks x 256 threads = 512 waves x 4 tiles = 2048 tiles.
// ---------------------------------------------------------------------------
__global__ void qk_apply_group(float* __restrict__ re, float* __restrict__ im,
                               const float* __restrict__ gate, int p) {
    int lane   = threadIdx.x & 31;
    int waveId = blockIdx.x * (blockDim.x >> 5) + (threadIdx.x >> 5); // 0..511
    int n    = lane & 15;        // column within tile / A row M
    int half = lane >> 4;        // 0 or 1
    unsigned maskLow = (1u << p) - 1u;

    // A (gate) per ISA A-layout: a.x = A[m][4c4 + 2*half], a.y = next column.
    v2f a[4];
#pragma unroll
    for (int c4 = 0; c4 < 4; ++c4) {
        int kb = 4 * c4 + half * 2;
        a[c4].x = gate[n * 16 + kb];
        a[c4].y = gate[n * 16 + kb + 1];
    }

#pragma unroll
    for (int t = 0; t < TILES_PER_WAVE; ++t) {
        int tile = waveId * TILES_PER_WAVE + t;
        unsigned j        = (unsigned)(tile * 16 + n);   // outer index 0..32767
        unsigned baseHigh = (j >> p) << (p + 3);
        unsigned baseLow  = j & maskLow;

        v8f acc = {};
#pragma unroll
        for (int c4 = 0; c4 < 4; ++c4) {
            int rb = 4 * c4 + half * 2;    // B row pair (rb, rb+1); no 7->8 cross
            const float* p0 = (rb & 8) ? im : re;
            const float* p1 = ((rb + 1) & 8) ? im : re;
            unsigned a0 = baseHigh | ((unsigned)(rb & 7) << p) | baseLow;
            unsigned a1 = baseHigh | ((unsigned)((rb + 1) & 7) << p) | baseLow;
            v2f b;
            b.x = p0[a0];
            b.y = p1[a1];
            acc = __builtin_amdgcn_wmma_f32_16x16x4_f32(
                /*neg_a=*/false, a[c4], /*neg_b=*/false, b,
                /*c_mod=*/(short)0, acc, /*reuse_a=*/false, /*reuse_b=*/false);
        }

        // D per ISA C/D layout: VGPR v holds row (v + 8*half), column n.
#pragma unroll
        for (int v = 0; v < 8; ++v) {
            int mo = v + half * 8;
            float* pd = (mo & 8) ? im : re;
            unsigned ad = baseHigh | ((unsigned)(mo & 7) << p) | baseLow;
            pd[ad] = acc[v];
        }
    }
}

// ---------------------------------------------------------------------------
// CNOT chain (w -> w+1 for w = 0..16) as one permutation:
// output bit j = XOR of input bits j..17 (suffix parity / inverse Gray code).
// Only needed between layers; the final chain is fused into qk_expvals.
// ---------------------------------------------------------------------------
__global__ void qk_cnot_chain(const float* __restrict__ sre, const float* __restrict__ sim,
                              float* __restrict__ dre, float* __restrict__ dim_) {
    unsigned i = blockIdx.x * blockDim.x + threadIdx.x;
    unsigned y = i;
    y ^= y >> 1; y ^= y >> 2; y ^= y >> 4; y ^= y >> 8; y ^= y >> 16;
    dre[y] = sre[i];
    dim_[y] = sim[i];
}

// ---------------------------------------------------------------------------
// <Z_w> on the post-CNOT state, with the final permutation fused:
// new[y(i)] = old[i]  =>  sum_j p_new[j]*sign_w(j) = sum_i p_old[i]*sign_w(y(i)).
// Register accumulators + LDS tree reduce + one atomicAdd per wire per block.
// ---------------------------------------------------------------------------
__global__ void qk_expvals(const float* __restrict__ re, const float* __restrict__ im,
                           float* __restrict__ expvals) {
    __shared__ float sdata[256];
    unsigned tid = threadIdx.x;
    float acc[NW];
#pragma unroll
    for (int w = 0; w < NW; ++w) acc[w] = 0.0f;

    for (unsigned i = blockIdx.x * blockDim.x + tid; i < DIM;
         i += gridDim.x * blockDim.x) {
        float pr = re[i], pi = im[i];
        float pp = pr * pr + pi * pi;
        unsigned y = i;                       // fused CNOT-chain index remap
        y ^= y >> 1; y ^= y >> 2; y ^= y >> 4; y ^= y >> 8; y ^= y >> 16;
#pragma unroll
        for (int w = 0; w < NW; ++w) {
            unsigned bit = (y >> (NW - 1 - w)) & 1u;
            acc[w] += bit ? -pp : pp;
        }
    }

    for (int w = 0; w < NW; ++w) {
        sdata[tid] = acc[w];
        __syncthreads();
        for (int s = 128; s > 0; s >>= 1) {
            if (tid < (unsigned)s) sdata[tid] += sdata[tid + s];
            __syncthreads();
        }
        if (tid == 0) atomicAdd(&expvals[w], sdata[0]);
        __syncthreads();
    }
}

// ---------------------------------------------------------------------------
// out[b] = dot(expvals, head_w) + head_b, broadcast to 128 batch elements.
// ---------------------------------------------------------------------------
__global__ void qk_head(const float* __restrict__ expvals,
                        const float* __restrict__ hw, const float* __restrict__ hb,
                        float* __restrict__ out, int batch) {
    int t = blockIdx.x * blockDim.x + threadIdx.x;
    if (t >= batch) return;
    float s = hb[0];
#pragma unroll
    for (int w = 0; w < NW; ++w) s += expvals[w] * hw[w];
    out[t] = s;
}

// ---------------------------------------------------------------------------
extern "C" void kernel_launch(void* const* d_in, const int* in_sizes, int n_in,
                              void* d_out, int out_size, void* d_ws, size_t ws_size,
                              hipStream_t stream) {
    (void)in_sizes; (void)n_in; (void)ws_size;
    // inputs: 0=state_real (ignored), 1=state_imag (ignored), 2=params,
    //         3=head_w, 4=head_b
    const float* params = (const float*)d_in[2];
    const float* head_w = (const float*)d_in[3];
    const float* head_b = (const float*)d_in[4];
    float* out = (float*)d_out;

    float* ws      = (float*)d_ws;
    float* Are     = ws;                 // DIM
    float* Aim     = ws + DIM;           // DIM
    float* Bre     = ws + 2 * DIM;       // DIM
    float* Bim     = ws + 3 * DIM;       // DIM
    float* gates   = ws + 4 * DIM;       // 12 * 256
    float* expvals = gates + 12 * 256;   // NW

    qk_init<<<DIM / 256, 256, 0, stream>>>(Are, Aim, expvals);
    qk_build_gates<<<12, 256, 0, stream>>>(params, gates);

    // layer 0: fused 3-wire WMMA gates (groups overlap in memory ->
    // sequential launches), then the fused CNOT-chain permutation A -> B.
    for (int g = 0; g < GROUPS; ++g)
        qk_apply_group<<<TILES / (8 * TILES_PER_WAVE), 256, 0, stream>>>(
            Are, Aim, gates + g * 256, 15 - 3 * g);
    qk_cnot_chain<<<DIM / 256, 256, 0, stream>>>(Are, Aim, Bre, Bim);

    // layer 1 on buffer B; its trailing CNOT chain is fused into qk_expvals.
    for (int g = 0; g < GROUPS; ++g)
        qk_apply_group<<<TILES / (8 * TILES_PER_WAVE), 256, 0, stream>>>(
            Bre, Bim, gates + (GROUPS + g) * 256, 15 - 3 * g);

    qk_expvals<<<256, 256, 0, stream>>>(Bre, Bim, expvals);
    qk_head<<<1, 128, 0, stream>>>(expvals, head_w, head_b, out, out_size);
}